// TextEmbedding_64630667870533
// MI455X (gfx1250) — compile-verified
//
#include <hip/hip_runtime.h>

typedef float v2f __attribute__((ext_vector_type(2)));
typedef float v8f __attribute__((ext_vector_type(8)));

#define DIMD 256
#define NW   32
#define LNEPS 1e-12f

// One workgroup per (b,s) token position: 32 words x 256 dims.
// Phase 1: gather + LayerNorm-center/scale each word row into LDS (f32, exact).
// Phase 2: sum over the 32 words with v_wmma_f32_16x16x4_f32 (A == ones, so
//          D[m,n] = sum_k B[k,n] regardless of exact K ordering in VGPRs),
//          then apply gamma/beta (which distribute over the word sum).
__global__ __launch_bounds__(128) void emb_ln_sum_kernel(
    const int* __restrict__ ids,      // [4096, 32]
    const float* __restrict__ table,  // [32000, 256]
    const float* __restrict__ gamma,  // [256]
    const float* __restrict__ beta,   // [256]
    float* __restrict__ out)          // [4096, 256]
{
    __shared__ float smem[NW * DIMD];     // 32 KB: normalized rows, row-major [w][d]

    const int t    = blockIdx.x;          // token position (b*128 + s)
    const int lane = threadIdx.x & 31;
    const int wave = threadIdx.x >> 5;    // 4 waves
    const int* idrow = ids + t * NW;

    // ---------------- Phase 1: per-word LayerNorm into LDS ----------------
    for (int i = 0; i < 8; ++i) {
        const int w   = wave * 8 + i;     // uniform per wave -> scalar load of idrow[w]
        const int row = idrow[w];
        const float* rp = table + (size_t)row * DIMD;

        // full 1 KB row coalesced across the wave: 32 lanes x 32 B
        const float4 x0 = *(const float4*)(rp + lane * 8);
        const float4 x1 = *(const float4*)(rp + lane * 8 + 4);

        // prefetch the next gathered row (global_prefetch_b8)
        if (i < 7) {
            const float* np = table + (size_t)idrow[w + 1] * DIMD;
            __builtin_prefetch(np + lane * 8, 0, 0);
        }

        float s  = x0.x + x0.y + x0.z + x0.w + x1.x + x1.y + x1.z + x1.w;
        float sq = x0.x*x0.x + x0.y*x0.y + x0.z*x0.z + x0.w*x0.w
                 + x1.x*x1.x + x1.y*x1.y + x1.z*x1.z + x1.w*x1.w;

        // wave32 butterfly reduction
        #pragma unroll
        for (int off = 16; off >= 1; off >>= 1) {
            s  += __shfl_xor(s,  off, 32);
            sq += __shfl_xor(sq, off, 32);
        }

        const float mu  = s  * (1.0f / DIMD);
        const float var = sq * (1.0f / DIMD) - mu * mu;
        const float r   = rsqrtf(var + LNEPS);

        float4 n0, n1;
        n0.x = (x0.x - mu) * r;  n0.y = (x0.y - mu) * r;
        n0.z = (x0.z - mu) * r;  n0.w = (x0.w - mu) * r;
        n1.x = (x1.x - mu) * r;  n1.y = (x1.y - mu) * r;
        n1.z = (x1.z - mu) * r;  n1.w = (x1.w - mu) * r;

        *(float4*)(&smem[w * DIMD + lane * 8])     = n0;
        *(float4*)(&smem[w * DIMD + lane * 8 + 4]) = n1;
    }
    __syncthreads();

    // ---------------- Phase 2: WMMA word-sum ----------------
    // A = all ones (16x4) -> every row of D holds the column sums of B.
    v2f a; a.x = 1.0f; a.y = 1.0f;

    const int col  = lane & 15;           // B column index N
    const int half = (lane >> 4) & 1;     // lane half selects word parity

    for (int ti = 0; ti < 4; ++ti) {
        const int n0 = (wave * 4 + ti) * 16;   // 16-dim column tile
        v8f c = {};

        #pragma unroll
        for (int ch = 0; ch < 8; ++ch) {       // 8 chunks of K=4 words
            const int w0 = ch * 4;
            v2f b;
            b.x = smem[(w0 + half)     * DIMD + n0 + col];
            b.y = smem[(w0 + 2 + half) * DIMD + n0 + col];
            c = __builtin_amdgcn_wmma_f32_16x16x4_f32(
                    /*neg_a=*/false, a, /*neg_b=*/false, b,
                    /*c_mod=*/(short)0, c, /*reuse_a=*/false, /*reuse_b=*/false);
        }

        // D row M=0 lives in c[0] on lanes 0..15 (N = lane)
        if (lane < 16) {
            const int d = n0 + lane;
            out[t * DIMD + d] = gamma[d] * c[0] + 32.0f * beta[d];
        }
    }
}

extern "C" void kernel_launch(void* const* d_in, const int* in_sizes, int n_in,
                              void* d_out, int out_size, void* d_ws, size_t ws_size,
                              hipStream_t stream) {
    (void)in_sizes; (void)n_in; (void)d_ws; (void)ws_size; (void)out_size;
    const int*   ids   = (const int*)d_in[0];     // [32,128,32] int32
    const float* table = (const float*)d_in[1];   // [32000,256] f32
    const float* gamma = (const float*)d_in[2];   // [256] f32
    const float* beta  = (const float*)d_in[3];   // [256] f32
    float*       out   = (float*)d_out;           // [32,128,256] f32

    const int tokens = 32 * 128;                  // 4096 workgroups
    emb_ln_sum_kernel<<<tokens, 128, 0, stream>>>(ids, table, gamma, beta, out);
}